// SegmentationBranch_57698590654975
// MI455X (gfx1250) — compile-verified
//
#include <hip/hip_runtime.h>
#include <math.h>

typedef __attribute__((ext_vector_type(2))) float v2f;
typedef __attribute__((ext_vector_type(4))) float v4f;
typedef __attribute__((ext_vector_type(8))) float v8f;

#define BATCH 4
#define CIN   512
#define HID   64
#define HB    60
#define WB    80
#define NPIXB (HB*WB)     // 4800 base pixels
#define HS    30
#define WS    40
#define NPIXS (HS*WS)     // 1200 half-res pixels
#define HF    480
#define WF    640
#define NCLS  11          // NUM_CLASSES + 1
#define PROB_N ((size_t)BATCH*NCLS*HF*WF)   // 13,516,800
#define SEG_N  ((size_t)BATCH*HF*WF)        // 1,228,800

// One wave computes a 16(M) x 16(N) f32 tile of  A[M,K] @ X[K,N]  using
// V_WMMA_F32_16X16X4_F32 (K=4 per op). Fragment layouts per CDNA5 ISA 7.12.2:
//   A 16x4 : lanes 0-15 hold M=lane, VGPR{0,1}=K{0,1}; lanes 16-31 K{2,3}
//   B 4x16 : lanes 0-15 hold N=lane, VGPR{0,1}=K{0,1}; lanes 16-31 K{2,3}
//   C 16x16: VGPR r -> M=r (lanes 0-15) / M=r+8 (lanes 16-31), N=lane&15
__device__ inline v8f gemm_tile_f32(const float* __restrict__ A, int lda, int arows,
                                    const float* __restrict__ X, int ldx,
                                    int m0, int n0, int K, int lane) {
  v8f acc = {};
  const int half = lane >> 4;      // 0 or 1
  const int l16  = lane & 15;
  const int arow = m0 + l16;
  const int ncol = n0 + l16;
  const bool avalid = arow < arows;
  const float* __restrict__ Arow = A + (size_t)arow * lda;
  #pragma unroll 4
  for (int k = 0; k < K; k += 4) {
    const int kk = k + half * 2;
    v2f a = {0.f, 0.f};
    if (avalid) a = *(const v2f*)(Arow + kk);            // 8B-aligned (kk even)
    v2f b;
    b.x = X[(size_t)kk * ldx + ncol];
    b.y = X[(size_t)(kk + 1) * ldx + ncol];
    acc = __builtin_amdgcn_wmma_f32_16x16x4_f32(
        /*neg_a=*/false, a, /*neg_b=*/false, b,
        /*c_mod=*/(short)0, acc, /*reuse_a=*/false, /*reuse_b=*/false);
  }
  return acc;
}

// -------- Kernel 1: feat2r = relu(W2 @ F2 + b2), (4,64,30,40) ----------------
// tiles: 4 batch * 4 Mtile * 75 Ntile = 1200 waves
__global__ void k_feat2(const float* __restrict__ f2, const float* __restrict__ w2,
                        const float* __restrict__ b2, float* __restrict__ feat2r) {
  const int wave = (blockIdx.x * blockDim.x + threadIdx.x) >> 5;
  const int lane = threadIdx.x & 31;
  const int nt = wave % 75;
  const int mt = (wave / 75) & 3;
  const int b  = wave / 300;
  const int m0 = mt * 16, n0 = nt * 16;
  const float* X = f2 + (size_t)b * CIN * NPIXS;
  v8f acc = gemm_tile_f32(w2, CIN, HID, X, NPIXS, m0, n0, CIN, lane);
  const int half = lane >> 4, n = n0 + (lane & 15);
  #pragma unroll
  for (int r = 0; r < 8; ++r) {
    const int m = m0 + r + half * 8;
    float v = acc[r] + b2[m];
    v = v > 0.f ? v : 0.f;
    feat2r[((size_t)b * HID + m) * NPIXS + n] = v;
  }
}

// -------- Kernel 2: combo = relu(W1 @ F1 + b1) + up2x(feat2r), (4,64,60,80) --
// tiles: 4 batch * 4 Mtile * 300 Ntile = 4800 waves
__global__ void k_combo(const float* __restrict__ f1, const float* __restrict__ w1,
                        const float* __restrict__ b1, const float* __restrict__ feat2r,
                        float* __restrict__ combo) {
  const int wave = (blockIdx.x * blockDim.x + threadIdx.x) >> 5;
  const int lane = threadIdx.x & 31;
  const int nt = wave % 300;
  const int mt = (wave / 300) & 3;
  const int b  = wave / 1200;
  const int m0 = mt * 16, n0 = nt * 16;
  const float* X = f1 + (size_t)b * CIN * NPIXB;
  v8f acc = gemm_tile_f32(w1, CIN, HID, X, NPIXB, m0, n0, CIN, lane);
  const int half = lane >> 4, n = n0 + (lane & 15);
  const int y = n / WB, x = n % WB;
  #pragma unroll
  for (int r = 0; r < 8; ++r) {
    const int m = m0 + r + half * 8;
    float v = acc[r] + b1[m];
    v = v > 0.f ? v : 0.f;
    v += feat2r[((size_t)b * HID + m) * NPIXS + (y >> 1) * WS + (x >> 1)];
    combo[((size_t)b * HID + m) * NPIXB + n] = v;
  }
}

// -------- Kernel 3: out GEMM + softmax + argmax + 8x replicated stores -------
// tiles: 4 batch * 300 Ntile = 1200 waves; each 16-pixel tile sits in one row y
__global__ void k_out(const float* __restrict__ combo, const float* __restrict__ wo,
                      const float* __restrict__ bo, float* __restrict__ prob,
                      float* __restrict__ seg, int* __restrict__ seg_base) {
  __shared__ float s_o[4][16][16];   // [wave][pixel][channel] pre-softmax (relu'd)
  __shared__ float s_p[4][16][12];   // [wave][pixel][channel] softmax probs
  __shared__ int   s_s[4][16];       // [wave][pixel] argmax label
  const int wv   = threadIdx.x >> 5;
  const int lane = threadIdx.x & 31;
  const int wave = blockIdx.x * 4 + wv;
  const int nt = wave % 300;
  const int b  = wave / 300;
  const int n0 = nt * 16;
  const float* X = combo + (size_t)b * HID * NPIXB;
  v8f acc = gemm_tile_f32(wo, HID, NCLS, X, NPIXB, 0, n0, HID, lane);
  const int half = lane >> 4, l16 = lane & 15;
  if (half == 0) {
    #pragma unroll
    for (int r = 0; r < 8; ++r) {
      float v = acc[r] + bo[r];
      s_o[wv][l16][r] = v > 0.f ? v : 0.f;
    }
  } else {
    #pragma unroll
    for (int r = 0; r < 3; ++r) {
      float v = acc[r] + bo[8 + r];
      s_o[wv][l16][8 + r] = v > 0.f ? v : 0.f;
    }
  }
  __syncthreads();
  if (half == 0) {
    float o[NCLS];
    float mx = -1e30f;
    int best = 0;
    #pragma unroll
    for (int c = 0; c < NCLS; ++c) {
      o[c] = s_o[wv][l16][c];
      if (o[c] > mx) { mx = o[c]; best = c; }   // strictly-greater => first max
    }
    float sum = 0.f;
    #pragma unroll
    for (int c = 0; c < NCLS; ++c) { o[c] = expf(o[c] - mx); sum += o[c]; }
    const float inv = 1.f / sum;
    #pragma unroll
    for (int c = 0; c < NCLS; ++c) s_p[wv][l16][c] = o[c] * inv;
    s_s[wv][l16] = best;
    seg_base[b * NPIXB + n0 + l16] = best;      // base-res labels for bbox pass
  }
  __syncthreads();
  // Replicated coalesced stores: each float4 lies inside one 8-wide block,
  // so all 4 elements come from base pixel p = x0 + (lane>>1).
  const int y  = n0 / WB;
  const int x0 = n0 % WB;
  const int p  = lane >> 1;
  const int Xs = x0 * 8 + lane * 4;
  #pragma unroll
  for (int c = 0; c < NCLS; ++c) {
    const float v = s_p[wv][p][c];
    const v4f vv = {v, v, v, v};
    float* dst = prob + (((size_t)b * NCLS + c) * HF + (size_t)y * 8) * WF + Xs;
    #pragma unroll
    for (int dy = 0; dy < 8; ++dy) *(v4f*)(dst + (size_t)dy * WF) = vv;
  }
  const float sv = (float)s_s[wv][p];
  const v4f sv4 = {sv, sv, sv, sv};
  float* sdst = seg + ((size_t)b * HF + (size_t)y * 8) * WF + Xs;
  #pragma unroll
  for (int dy = 0; dy < 8; ++dy) *(v4f*)(sdst + (size_t)dy * WF) = sv4;
}

// -------- Kernel 4: bbox reduction on base-res labels, rescale x8 ------------
// 36 blocks = 4 batches * 9 classes (cls 1..9), 256 threads each
__global__ void k_bbx(const int* __restrict__ seg_base, float* __restrict__ bbx) {
  __shared__ int s_cnt[256], s_xmin[256], s_xmax[256], s_ymin[256], s_ymax[256];
  const int b   = blockIdx.x / 9;
  const int cls = (blockIdx.x % 9) + 1;
  const int tid = threadIdx.x;
  int cnt = 0, xmn = 1 << 30, xmx = -1, ymn = 1 << 30, ymx = -1;
  for (int i = tid; i < NPIXB; i += 256) {
    if (seg_base[b * NPIXB + i] == cls) {
      const int y = i / WB, x = i % WB;
      ++cnt;
      xmn = x < xmn ? x : xmn;  xmx = x > xmx ? x : xmx;
      ymn = y < ymn ? y : ymn;  ymx = y > ymx ? y : ymx;
    }
  }
  s_cnt[tid] = cnt; s_xmin[tid] = xmn; s_xmax[tid] = xmx;
  s_ymin[tid] = ymn; s_ymax[tid] = ymx;
  __syncthreads();
  for (int s = 128; s > 0; s >>= 1) {
    if (tid < s) {
      s_cnt[tid] += s_cnt[tid + s];
      if (s_xmin[tid + s] < s_xmin[tid]) s_xmin[tid] = s_xmin[tid + s];
      if (s_xmax[tid + s] > s_xmax[tid]) s_xmax[tid] = s_xmax[tid + s];
      if (s_ymin[tid + s] < s_ymin[tid]) s_ymin[tid] = s_ymin[tid + s];
      if (s_ymax[tid + s] > s_ymax[tid]) s_ymax[tid] = s_ymax[tid + s];
    }
    __syncthreads();
  }
  if (tid == 0) {
    // full-res count = 64 * base count; threshold 500 -> base count >= 8
    const bool valid = s_cnt[0] * 64 >= 500;
    float* row = bbx + (size_t)blockIdx.x * 6;
    if (valid) {
      row[0] = (float)b;
      row[1] = (float)(8 * s_xmin[0]);
      row[2] = (float)(8 * s_ymin[0]);
      row[3] = (float)(8 * s_xmax[0] + 7);
      row[4] = (float)(8 * s_ymax[0] + 7);
      row[5] = (float)cls;
    } else {
      for (int j = 0; j < 6; ++j) row[j] = -1.f;
    }
  }
}

extern "C" void kernel_launch(void* const* d_in, const int* in_sizes, int n_in,
                              void* d_out, int out_size, void* d_ws, size_t ws_size,
                              hipStream_t stream) {
  const float* feature1 = (const float*)d_in[0];
  const float* feature2 = (const float*)d_in[1];
  const float* w1 = (const float*)d_in[2];
  const float* b1 = (const float*)d_in[3];
  const float* w2 = (const float*)d_in[4];
  const float* b2 = (const float*)d_in[5];
  const float* wo = (const float*)d_in[6];
  const float* bo = (const float*)d_in[7];

  float* out  = (float*)d_out;
  float* prob = out;                       // (4,11,480,640)
  float* seg  = out + PROB_N;              // (4,480,640) as float
  float* bbx  = out + PROB_N + SEG_N;      // (36,6) as float

  float* feat2r   = (float*)d_ws;                          // 4*64*1200 floats
  float* combo    = feat2r + (size_t)BATCH * HID * NPIXS;  // 4*64*4800 floats
  int*   seg_base = (int*)(combo + (size_t)BATCH * HID * NPIXB); // 4*4800 ints

  k_feat2<<<300, 128, 0, stream>>>(feature2, w2, b2, feat2r);          // 1200 waves
  k_combo<<<1200, 128, 0, stream>>>(feature1, w1, b1, feat2r, combo);  // 4800 waves
  k_out  <<<300, 128, 0, stream>>>(combo, wo, bo, prob, seg, seg_base);// 1200 waves
  k_bbx  <<<36, 256, 0, stream>>>(seg_base, bbx);
}